// MoPro_39659728011353
// MI455X (gfx1250) — compile-verified
//
#include <hip/hip_runtime.h>
#include <hip/hip_bf16.h>

typedef __attribute__((ext_vector_type(2))) float v2f;
typedef __attribute__((ext_vector_type(8))) float v8f;

#define NB   1024      // batch
#define ND   128       // feature dim
#define NK   65536     // queue length
#define NC   1000      // classes
#define LDL  (NK + 1)  // logits row stride = 65537
#define INV_T 10.0f

// output offsets (flat fp32 elements, in return order)
#define OFF_LOGITS   ((size_t)0)
#define OFF_PROTO_L  ((size_t)NB * LDL)                 // 67,109,888
#define OFF_QUEUE    (OFF_PROTO_L + (size_t)NB * NC)    // 68,133,888
#define OFF_PROTOS   (OFF_QUEUE + (size_t)ND * NK)      // 76,522,496

// ---------------------------------------------------------------------------
// l_neg = (q @ queue) / T   -> logits[:, 1:]
// wave tile: 16 (M) x 64 (N), fp32 WMMA 16x16x4, K loop over 128
// ---------------------------------------------------------------------------
__global__ __launch_bounds__(128) void neg_gemm(const float* __restrict__ q,
                                                const float* __restrict__ queue,
                                                float* __restrict__ out) {
    const int lane = threadIdx.x & 31;
    const int wave = threadIdx.x >> 5;
    const int n0 = (blockIdx.x * 4 + wave) * 64;
    const int m0 = blockIdx.y * 16;
    const int half = lane >> 4;        // 0: K pair {0,1}, 1: K pair {2,3}
    const int kb   = half * 2;
    const int l16  = lane & 15;

    const float* qrow = q + (size_t)(m0 + l16) * ND + kb;

    v8f acc0 = {}, acc1 = {}, acc2 = {}, acc3 = {};

    for (int k = 0; k < ND; k += 4) {
        v2f a = *(const v2f*)(qrow + k);
        const float* bp = queue + (size_t)(k + kb) * NK + n0 + l16;
        v2f b0, b1, b2, b3;
        b0.x = bp[0];  b0.y = bp[NK + 0];
        b1.x = bp[16]; b1.y = bp[NK + 16];
        b2.x = bp[32]; b2.y = bp[NK + 32];
        b3.x = bp[48]; b3.y = bp[NK + 48];
        acc0 = __builtin_amdgcn_wmma_f32_16x16x4_f32(false, a, false, b0, (short)0, acc0, false, false);
        acc1 = __builtin_amdgcn_wmma_f32_16x16x4_f32(false, a, false, b1, (short)0, acc1, false, false);
        acc2 = __builtin_amdgcn_wmma_f32_16x16x4_f32(false, a, false, b2, (short)0, acc2, false, false);
        acc3 = __builtin_amdgcn_wmma_f32_16x16x4_f32(false, a, false, b3, (short)0, acc3, false, false);
    }

#pragma unroll
    for (int r = 0; r < 8; ++r) {
        size_t base = (size_t)(m0 + r + 8 * half) * LDL + 1 + n0 + l16;
        out[base +  0] = acc0[r] * INV_T;
        out[base + 16] = acc1[r] * INV_T;
        out[base + 32] = acc2[r] * INV_T;
        out[base + 48] = acc3[r] * INV_T;
    }
}

// ---------------------------------------------------------------------------
// logits_proto = (q @ prototypes^T) / T    [1024 x 1000]
// one wave per 16x16 tile; prototype rows give contiguous float2 B-fragments
// ---------------------------------------------------------------------------
__global__ __launch_bounds__(32) void proto_gemm(const float* __restrict__ q,
                                                 const float* __restrict__ protos,
                                                 float* __restrict__ out) {
    const int lane = threadIdx.x;
    const int n0 = blockIdx.x * 16;   // 63 tiles cover 1008, guarded at 1000
    const int m0 = blockIdx.y * 16;
    const int half = lane >> 4;
    const int kb   = half * 2;
    const int l16  = lane & 15;
    const int n    = n0 + l16;
    const int ncl  = n < NC ? n : NC - 1;   // clamp loads, keep EXEC uniform

    const float* qrow = q + (size_t)(m0 + l16) * ND + kb;
    const float* prow = protos + (size_t)ncl * ND + kb;

    v8f acc = {};
#pragma unroll 8
    for (int k = 0; k < ND; k += 4) {
        v2f a = *(const v2f*)(qrow + k);
        v2f b = *(const v2f*)(prow + k);
        acc = __builtin_amdgcn_wmma_f32_16x16x4_f32(false, a, false, b, (short)0, acc, false, false);
    }

    if (n < NC) {
#pragma unroll
        for (int r = 0; r < 8; ++r)
            out[(size_t)(m0 + r + 8 * half) * NC + n] = acc[r] * INV_T;
    }
}

// ---------------------------------------------------------------------------
// l_pos[b] = dot(q[b], k[b]) / T  -> logits[:, 0];  one wave per row
// ---------------------------------------------------------------------------
__global__ __launch_bounds__(256) void pos_kernel(const float* __restrict__ q,
                                                  const float* __restrict__ kk,
                                                  float* __restrict__ out) {
    int gid  = blockIdx.x * 256 + threadIdx.x;
    int row  = gid >> 5;
    int lane = gid & 31;
    const float4* qr = (const float4*)(q + (size_t)row * ND);
    const float4* kr = (const float4*)(kk + (size_t)row * ND);
    float4 a = qr[lane];
    float4 b = kr[lane];
    float s = a.x * b.x + a.y * b.y + a.z * b.z + a.w * b.w;
#pragma unroll
    for (int off = 16; off > 0; off >>= 1) s += __shfl_xor(s, off, 32);
    if (lane == 0) out[(size_t)row * LDL] = s * INV_T;
}

// ---------------------------------------------------------------------------
// new_queue: queue with columns [0,1024) replaced by k^T; float4 streaming
// ---------------------------------------------------------------------------
__global__ __launch_bounds__(256) void queue_copy(const float* __restrict__ kk,
                                                  const float* __restrict__ queue,
                                                  float* __restrict__ out) {
    int tid = blockIdx.x * 256 + threadIdx.x;   // ND * NK/4 threads
    int d   = tid >> 14;                        // NK/4 = 16384 float4 per row
    int n   = (tid & 16383) * 4;
    size_t off = (size_t)d * NK + n;
    float4 v;
    if (n >= NB) {
        v = *(const float4*)(queue + off);
    } else {
        v.x = kk[(size_t)(n + 0) * ND + d];
        v.y = kk[(size_t)(n + 1) * ND + d];
        v.z = kk[(size_t)(n + 2) * ND + d];
        v.w = kk[(size_t)(n + 3) * ND + d];
    }
    *(float4*)(out + off) = v;
}

// ---------------------------------------------------------------------------
// new_prototypes: per-class sequential EMA (deterministic) + L2 norm
// grid = NC blocks, block = ND threads
// ---------------------------------------------------------------------------
__global__ __launch_bounds__(128) void proto_ema(const float* __restrict__ q,
                                                 const float* __restrict__ protos,
                                                 const int* __restrict__ target,
                                                 float* __restrict__ out) {
    const int c = blockIdx.x;
    const int d = threadIdx.x;
    __shared__ int tgt[NB];
    for (int i = d; i < NB; i += ND) tgt[i] = target[i];
    __syncthreads();

    int n_c = 0;
    for (int i = 0; i < NB; ++i) n_c += (tgt[i] == c);

    const float log2m = -0.0014434742f;   // log2(0.999)
    float acc = 0.0f;
    int j = 0;
    for (int i = 0; i < NB; ++i) {
        if (tgt[i] == c) {
            ++j;
            float w = 0.001f * exp2f((float)(n_c - j) * log2m);
            acc += w * q[(size_t)i * ND + d];
        }
    }
    float val = protos[(size_t)c * ND + d] * exp2f((float)n_c * log2m) + acc;

    __shared__ float red[ND];
    red[d] = val * val;
    __syncthreads();
#pragma unroll
    for (int s = ND / 2; s > 0; s >>= 1) {
        if (d < s) red[d] += red[d + s];
        __syncthreads();
    }
    float norm = fmaxf(sqrtf(red[0]), 1e-12f);
    out[(size_t)c * ND + d] = val / norm;
}

// ---------------------------------------------------------------------------
extern "C" void kernel_launch(void* const* d_in, const int* in_sizes, int n_in,
                              void* d_out, int out_size, void* d_ws, size_t ws_size,
                              hipStream_t stream) {
    const float* q      = (const float*)d_in[0];
    const float* k      = (const float*)d_in[1];
    const float* queue  = (const float*)d_in[2];
    const float* protos = (const float*)d_in[3];
    const int*   target = (const int*)d_in[4];
    float* out = (float*)d_out;

    // l_neg: grid (NK/(4*64), NB/16) = (256, 64), 4 waves/block
    neg_gemm<<<dim3(NK / 256, NB / 16), 128, 0, stream>>>(q, queue, out + OFF_LOGITS);

    // logits_proto: one wave per 16x16 tile
    proto_gemm<<<dim3((NC + 15) / 16, NB / 16), 32, 0, stream>>>(q, protos, out + OFF_PROTO_L);

    // l_pos: 1024 waves
    pos_kernel<<<dim3(NB * 32 / 256), 256, 0, stream>>>(q, k, out + OFF_LOGITS);

    // new_queue: ND*NK/4 float4 threads
    queue_copy<<<dim3(ND * NK / 4 / 256), 256, 0, stream>>>(k, queue, out + OFF_QUEUE);

    // new_prototypes
    proto_ema<<<dim3(NC), ND, 0, stream>>>(q, protos, target, out + OFF_PROTOS);
}